// QuantumLayer_65481071405362
// MI455X (gfx1250) — compile-verified
//
#include <hip/hip_runtime.h>

// ---------------------------------------------------------------------------
// 20-qubit state-vector simulator for MI455X (gfx1250, wave32), fused schedule.
//
// ψ: 2^20 complex64 (float2) = 8 MB, lives in d_ws (fits in 192 MB L2).
// Qubit q == flat bit (19-q) of the state index.
//
// Schedule (12 launches, ~128 MB L2 traffic total):
//   * per gate layer (RY / Rot): 2 LDS-tile passes over commuting 1q gates
//       - low pass : contiguous 2^14-amp tile, gates on flat bits 0..13
//       - high pass: strided tile (6 high bits x 256 low), bits 14..19
//     gates applied in LDS via V_WMMA_F32_16X16X4_F32 (complex 2x2 gate
//     embedded as a replicated real 4x4 in the A operand; 16 pairs/wave/WMMA)
//   * each 19-CNOT ladder == Gray-code relabeling: psi'[j] = psi[j ^ (j>>1)]
//     (one gather pass per layer instead of 19 swap passes)
//   * deterministic 2-stage reduction of sign(bit19)*|amp|^2
// Tile fill/drain uses GLOBAL_LOAD_ASYNC_TO_LDS_B128 /
// GLOBAL_STORE_ASYNC_FROM_LDS_B128 (+ s_wait_asynccnt) when available.
// ---------------------------------------------------------------------------

#define NQ        20
#define NSTATES   (1u << NQ)
#define NGATES    60
#define TILE_BITS 14
#define TILE      (1u << TILE_BITS)        // 16384 amplitudes = 128 KB LDS
#define TILE_F4   (TILE / 2u)              // 8192 float4 (2 amps each)
#define TPB       256u                     // 8 waves
#define R1BLOCKS  1024

typedef __attribute__((ext_vector_type(2))) float v2f;
typedef __attribute__((ext_vector_type(8))) float v8f;
typedef int b128 __attribute__((vector_size(16)));   // builtin's pointee type

#if defined(__has_builtin)
#if __has_builtin(__builtin_amdgcn_wmma_f32_16x16x4_f32)
#define HAVE_WMMA_F32X4 1
#endif
#if __has_builtin(__builtin_amdgcn_global_load_async_to_lds_b128) && \
    __has_builtin(__builtin_amdgcn_global_store_async_from_lds_b128) && \
    __has_builtin(__builtin_amdgcn_s_wait_asynccnt)
#define HAVE_ASYNC_LDS 1
#endif
#endif

// Typed address-space casts (AS matched already; pointee must be b128 exactly).
#define ASG(p) ((__attribute__((address_space(1))) b128*)(unsigned long long)(p))
#define ASL(p) ((__attribute__((address_space(3))) b128*)(unsigned)(unsigned long long)(p))

// ---------------------------------------------------------------------------
// Gate matrices: 60 gates x 8 floats {g00r,g00i,g01r,g01i,g10r,g10i,g11r,g11i}
// slot q<20: RY(x[q]); slot 20+l*20+q: Rot(params[l,q,:]).
// ---------------------------------------------------------------------------
__global__ void qsim_compute_gates(const float* __restrict__ x,
                                   const float* __restrict__ params,
                                   float* __restrict__ gbuf) {
    int t = threadIdx.x;
    if (t < 20) {
        float h = 0.5f * x[t];
        float c = cosf(h), s = sinf(h);
        float* g = gbuf + t * 8;
        g[0] = c;  g[1] = 0.0f;
        g[2] = -s; g[3] = 0.0f;
        g[4] = s;  g[5] = 0.0f;
        g[6] = c;  g[7] = 0.0f;
    } else if (t < NGATES) {
        int i = t - 20;                       // l*20 + q
        const float* p = params + i * 3;
        float phi = p[0], th = p[1], om = p[2];
        float c = cosf(0.5f * th), s = sinf(0.5f * th);
        float a = 0.5f * (phi + om), b = 0.5f * (phi - om);
        float ca = cosf(a), sa = sinf(a), cb = cosf(b), sb = sinf(b);
        float* g = gbuf + t * 8;
        g[0] = c * ca;  g[1] = -c * sa;   // g00 = e^{-ia} c
        g[2] = -s * cb; g[3] = -s * sb;   // g01 = -e^{+ib} s
        g[4] = s * cb;  g[5] = -s * sb;   // g10 = e^{-ib} s
        g[6] = c * ca;  g[7] = c * sa;    // g11 = e^{+ia} c
    }
}

__global__ void qsim_init_state(float2* __restrict__ psi) {
    unsigned i = blockIdx.x * blockDim.x + threadIdx.x;
    float2 v;
    v.x = (i == 0u) ? 1.0f : 0.0f;
    v.y = 0.0f;
    psi[i] = v;
}

// Real 4x4 embedding M of the complex 2x2 gate, element M[r][k].
__device__ __forceinline__ float qsim_m4(const float* __restrict__ g,
                                         unsigned r, unsigned k) {
    unsigned ri = r >> 1, rc = r & 1u;
    unsigned kc = k >> 1, kk = k & 1u;
    float re = g[ri * 4u + kc * 2u + 0u];
    float im = g[ri * 4u + kc * 2u + 1u];
    return (rc == 0u) ? (kk ? -im : re) : (kk ? re : im);
}

// ---------------------------------------------------------------------------
// Apply one 1q gate on LDS-tile local bit b. 8 waves x 64 iterations x 16
// pairs = 2^13 pairs. Lanes 0-15 own a0 of pair (lane&15), lanes 16-31 own a1.
// EXEC stays all-ones (uniform loops, branch-free selects around the WMMA).
// ---------------------------------------------------------------------------
__device__ __forceinline__ void qsim_gate_lds(float2* tile,
                                              const float* __restrict__ g,
                                              int b, unsigned t) {
    unsigned lane = t & 31u, wv = t >> 5;
    unsigned lowm = (1u << b) - 1u;
#if HAVE_WMMA_F32X4
    unsigned r = lane & 3u;
    unsigned kbase = (lane < 16u) ? 0u : 2u;
    v2f A;
    A.x = qsim_m4(g, r, kbase + 0u);
    A.y = qsim_m4(g, r, kbase + 1u);
#pragma unroll 4
    for (unsigned it = 0; it < 64u; ++it) {
        unsigned p   = wv * 1024u + it * 16u + (lane & 15u);
        unsigned i0  = ((p >> b) << (b + 1)) | (p & lowm);
        unsigned idx = (lane < 16u) ? i0 : (i0 | (1u << b));
        float2 amp = tile[idx];
        v2f B;
        B.x = amp.x;
        B.y = amp.y;
        v8f C = {};
        v8f D = __builtin_amdgcn_wmma_f32_16x16x4_f32(
            false, A, false, B, (short)0, C, false, false);
        float2 o;
        o.x = (lane < 16u) ? D[0] : D[2];
        o.y = (lane < 16u) ? D[1] : D[3];
        tile[idx] = o;
    }
#else
    if (lane < 16u) {
        for (unsigned it = 0; it < 64u; ++it) {
            unsigned p  = wv * 1024u + it * 16u + lane;
            unsigned i0 = ((p >> b) << (b + 1)) | (p & lowm);
            unsigned i1 = i0 | (1u << b);
            float2 a0 = tile[i0], a1 = tile[i1];
            float2 o0, o1;
            o0.x = g[0] * a0.x - g[1] * a0.y + g[2] * a1.x - g[3] * a1.y;
            o0.y = g[0] * a0.y + g[1] * a0.x + g[2] * a1.y + g[3] * a1.x;
            o1.x = g[4] * a0.x - g[5] * a0.y + g[6] * a1.x - g[7] * a1.y;
            o1.y = g[4] * a0.y + g[5] * a0.x + g[6] * a1.y + g[7] * a1.x;
            tile[i0] = o0;
            tile[i1] = o1;
        }
    }
#endif
}

// ---------------------------------------------------------------------------
// Low pass: contiguous tile [blk*2^14, (blk+1)*2^14); gates on bits 0..13
// (qubits 6..19). 64 blocks.
// ---------------------------------------------------------------------------
__global__ void __launch_bounds__(TPB)
qsim_fused_low(float2* __restrict__ psi, const float* __restrict__ gbuf,
               int slotBase) {
    __shared__ __align__(16) float2 tile[TILE];
    unsigned t = threadIdx.x;
    float4* gsrc  = (float4*)(psi + ((unsigned)blockIdx.x << TILE_BITS));
    float4* ltile = (float4*)tile;

#if HAVE_ASYNC_LDS
    for (unsigned i = 0; i < TILE_F4 / TPB; ++i) {
        unsigned f = i * TPB + t;
        __builtin_amdgcn_global_load_async_to_lds_b128(ASG(gsrc + f),
                                                       ASL(ltile + f), 0, 0);
    }
    __builtin_amdgcn_s_wait_asynccnt(0);
#else
    for (unsigned i = 0; i < TILE_F4 / TPB; ++i) {
        unsigned f = i * TPB + t;
        ltile[f] = gsrc[f];
    }
#endif
    __syncthreads();

    for (int qq = 6; qq < 20; ++qq) {                // bits 13..0
        qsim_gate_lds(tile, gbuf + (slotBase + qq) * 8, 19 - qq, t);
        __syncthreads();
    }

#if HAVE_ASYNC_LDS
    for (unsigned i = 0; i < TILE_F4 / TPB; ++i) {
        unsigned f = i * TPB + t;
        __builtin_amdgcn_global_store_async_from_lds_b128(ASG(gsrc + f),
                                                          ASL(ltile + f), 0, 0);
    }
    __builtin_amdgcn_s_wait_asynccnt(0);
#else
    for (unsigned i = 0; i < TILE_F4 / TPB; ++i) {
        unsigned f = i * TPB + t;
        gsrc[f] = ltile[f];
    }
#endif
}

// ---------------------------------------------------------------------------
// High pass: tile = { (h<<14) | (blk<<8) | r : h in [0,64), r in [0,256) };
// local amp index (h<<8)|r, so global bit 14+k == local bit 8+k.
// Gates on bits 14..19 (qubits 0..5). 64 blocks.
// ---------------------------------------------------------------------------
__global__ void __launch_bounds__(TPB)
qsim_fused_high(float2* __restrict__ psi, const float* __restrict__ gbuf,
                int slotBase) {
    __shared__ __align__(16) float2 tile[TILE];
    unsigned t = threadIdx.x;
    unsigned w = blockIdx.x;                          // occupies bits 8..13
    float4* ltile = (float4*)tile;

#if HAVE_ASYNC_LDS
    for (unsigned i = 0; i < TILE_F4 / TPB; ++i) {
        unsigned f = i * TPB + t;                     // local float4 index
        unsigned h = f >> 7, rp = f & 127u;
        float4* gp = (float4*)(psi + ((h << 14) | (w << 8) | (rp << 1)));
        __builtin_amdgcn_global_load_async_to_lds_b128(ASG(gp),
                                                       ASL(ltile + f), 0, 0);
    }
    __builtin_amdgcn_s_wait_asynccnt(0);
#else
    for (unsigned i = 0; i < TILE_F4 / TPB; ++i) {
        unsigned f = i * TPB + t;
        unsigned h = f >> 7, rp = f & 127u;
        ltile[f] = *(float4*)(psi + ((h << 14) | (w << 8) | (rp << 1)));
    }
#endif
    __syncthreads();

    for (int qq = 0; qq < 6; ++qq) {                 // local bits 13..8
        qsim_gate_lds(tile, gbuf + (slotBase + qq) * 8, 13 - qq, t);
        __syncthreads();
    }

#if HAVE_ASYNC_LDS
    for (unsigned i = 0; i < TILE_F4 / TPB; ++i) {
        unsigned f = i * TPB + t;
        unsigned h = f >> 7, rp = f & 127u;
        float4* gp = (float4*)(psi + ((h << 14) | (w << 8) | (rp << 1)));
        __builtin_amdgcn_global_store_async_from_lds_b128(ASG(gp),
                                                          ASL(ltile + f), 0, 0);
    }
    __builtin_amdgcn_s_wait_asynccnt(0);
#else
    for (unsigned i = 0; i < TILE_F4 / TPB; ++i) {
        unsigned f = i * TPB + t;
        unsigned h = f >> 7, rp = f & 127u;
        *(float4*)(psi + ((h << 14) | (w << 8) | (rp << 1))) = ltile[f];
    }
#endif
}

// ---------------------------------------------------------------------------
// CNOT ladder (q=0..18) == Gray-code relabeling: dst[j] = src[j ^ (j>>1)].
// ---------------------------------------------------------------------------
__global__ void qsim_gray(float2* __restrict__ dst,
                          const float2* __restrict__ src) {
    unsigned j = blockIdx.x * blockDim.x + threadIdx.x;
    dst[j] = src[j ^ (j >> 1)];
}

// ---------------------------------------------------------------------------
// Deterministic two-stage reduction of sign(bit19)*|amp|^2.
// ---------------------------------------------------------------------------
__global__ void qsim_reduce1(const float2* __restrict__ psi,
                             float* __restrict__ partial) {
    __shared__ float sm[256];
    unsigned t = threadIdx.x;
    unsigned base = blockIdx.x * 1024u;
    float acc = 0.0f;
#pragma unroll
    for (int it = 0; it < 4; ++it) {
        unsigned i = base + (unsigned)it * 256u + t;
        float2 a = psi[i];
        float p2 = a.x * a.x + a.y * a.y;
        acc += (i & (1u << 19)) ? -p2 : p2;
    }
    sm[t] = acc;
    __syncthreads();
    for (int s = 128; s > 0; s >>= 1) {
        if ((int)t < s) sm[t] += sm[t + s];
        __syncthreads();
    }
    if (t == 0) partial[blockIdx.x] = sm[0];
}

__global__ void qsim_reduce2(const float* __restrict__ partial,
                             float* __restrict__ out) {
    __shared__ float sm[256];
    unsigned t = threadIdx.x;
    float acc = partial[t] + partial[t + 256] + partial[t + 512] + partial[t + 768];
    sm[t] = acc;
    __syncthreads();
    for (int s = 128; s > 0; s >>= 1) {
        if ((int)t < s) sm[t] += sm[t + s];
        __syncthreads();
    }
    if (t == 0) out[0] = sm[0];
}

// ---------------------------------------------------------------------------
// Launch sequence (graph-capture safe: kernels only, all on `stream`).
// ---------------------------------------------------------------------------
extern "C" void kernel_launch(void* const* d_in, const int* in_sizes, int n_in,
                              void* d_out, int out_size, void* d_ws, size_t ws_size,
                              hipStream_t stream) {
    const float* x      = (const float*)d_in[0];   // [20]
    const float* params = (const float*)d_in[1];   // [2,20,3] flat

    // d_ws layout: gates @0 (2KB); psiA @4KB (8MB); psiB @4KB+8MB (8MB);
    // partials after.
    float*  gbuf    = (float*)d_ws;
    float2* psiA    = (float2*)((char*)d_ws + 4096);
    float2* psiB    = (float2*)((char*)d_ws + 4096 + sizeof(float2) * NSTATES);
    float*  partial = (float*)((char*)d_ws + 4096 + 2 * sizeof(float2) * NSTATES);
    float*  out     = (float*)d_out;

    qsim_compute_gates<<<1, 64, 0, stream>>>(x, params, gbuf);
    qsim_init_state<<<NSTATES / TPB, TPB, 0, stream>>>(psiA);

    const int TB = NSTATES / TILE;                 // 64 tile blocks

    // Angle-encoding RY layer (slots 0..19)
    qsim_fused_low <<<TB, TPB, 0, stream>>>(psiA, gbuf, 0);
    qsim_fused_high<<<TB, TPB, 0, stream>>>(psiA, gbuf, 0);

    // Layer 0: Rot (slots 20..39), then CNOT ladder (Gray relabel)
    qsim_fused_low <<<TB, TPB, 0, stream>>>(psiA, gbuf, 20);
    qsim_fused_high<<<TB, TPB, 0, stream>>>(psiA, gbuf, 20);
    qsim_gray<<<NSTATES / TPB, TPB, 0, stream>>>(psiB, psiA);

    // Layer 1: Rot (slots 40..59), then CNOT ladder
    qsim_fused_low <<<TB, TPB, 0, stream>>>(psiB, gbuf, 40);
    qsim_fused_high<<<TB, TPB, 0, stream>>>(psiB, gbuf, 40);
    qsim_gray<<<NSTATES / TPB, TPB, 0, stream>>>(psiA, psiB);

    qsim_reduce1<<<R1BLOCKS, TPB, 0, stream>>>(psiA, partial);
    qsim_reduce2<<<1, TPB, 0, stream>>>(partial, out);
}